// GCNCustom_42314017800850
// MI455X (gfx1250) — compile-verified
//
#include <hip/hip_runtime.h>

typedef float v2f __attribute__((ext_vector_type(2)));
typedef float v8f __attribute__((ext_vector_type(8)));

#define BM 128
#define BN 128
#define BK 32
#define LDA_S 34              // A tile row stride (floats), padded: 16 strided rows -> 16 distinct banks
#define LDB_P (2 * BN + 32)   // B pair-row stride (floats): +288 ≡ +32 banks -> k-halves use disjoint bank halves

// Computes a 128x128 tile of C = A[M,K] * B[K,N] (row-major), accumulating into acc[4][2]
// (each wave owns a 64x32 sub-tile = 4x2 f32 WMMA 16x16 fragments).
// B tile is stored k-pair-interleaved in LDS: Bs[k/2][n] = {B[k][n], B[k+1][n]} so a
// B fragment is one aligned ds_load_b64 (matches V_WMMA_F32_16X16X4_F32 operand layout).
__device__ __forceinline__ void gemm_mainloop(
    const float* __restrict__ A, const float* __restrict__ B,
    int lda, int ldb, int K,
    int tileM, int tileN,
    float* As, float* Bs,
    v8f acc[4][2])
{
    const int tid  = threadIdx.x;
    const int lane = tid & 31;
    const int wave = tid >> 5;      // 0..7
    const int wm   = wave >> 2;     // 0..1  -> 64-row half
    const int wn   = wave & 3;      // 0..3  -> 32-col quarter
    const int lr   = lane & 15;     // row (A) / col (B) within 16
    const int lh   = lane >> 4;     // k-half selector (K pairs {0,1} vs {2,3})

    // global->LDS cooperative load mapping (256 threads)
    const int a_row = tid >> 3;          // 0..31 (x4 strided by 32 -> 128 rows)
    const int a_col = (tid & 7) << 2;    // 0..28 (covers BK=32)
    const int b_kp  = tid >> 5;          // 0..7  (x2 strided by 8 -> 16 k-pairs)
    const int b_col = (tid & 31) << 2;   // 0..124 (covers BN=128)

    const float* Ag = A + (size_t)tileM * lda;
    const float* Bg = B + tileN;

    float4 areg[4];
    float4 breg0[2], breg1[2];

    auto load_tile = [&](int k0) {
        #pragma unroll
        for (int i = 0; i < 4; ++i)
            areg[i] = *(const float4*)(Ag + (size_t)(a_row + 32 * i) * lda + k0 + a_col);
        #pragma unroll
        for (int i = 0; i < 2; ++i) {
            const int k = k0 + 2 * (b_kp + 8 * i);
            breg0[i] = *(const float4*)(Bg + (size_t)(k)     * ldb + b_col);
            breg1[i] = *(const float4*)(Bg + (size_t)(k + 1) * ldb + b_col);
        }
    };

    auto store_tile = [&]() {
        #pragma unroll
        for (int i = 0; i < 4; ++i) {
            float* dst = As + (a_row + 32 * i) * LDA_S + a_col;
            dst[0] = areg[i].x; dst[1] = areg[i].y; dst[2] = areg[i].z; dst[3] = areg[i].w;
        }
        #pragma unroll
        for (int i = 0; i < 2; ++i) {
            // interleave rows k, k+1 as float2 pairs per column
            float* dst = Bs + (b_kp + 8 * i) * LDB_P + 2 * b_col;
            dst[0] = breg0[i].x; dst[1] = breg1[i].x;
            dst[2] = breg0[i].y; dst[3] = breg1[i].y;
            dst[4] = breg0[i].z; dst[5] = breg1[i].z;
            dst[6] = breg0[i].w; dst[7] = breg1[i].w;
        }
    };

    load_tile(0);
    store_tile();
    __syncthreads();

    for (int k0 = 0; k0 < K; k0 += BK) {
        const bool has_next = (k0 + BK) < K;
        if (has_next) {
            // register-staged prefetch of next K-tile (overlaps the WMMA loop below)
            load_tile(k0 + BK);
            if (k0 + 2 * BK < K) {
                // warm caches two tiles ahead (global_prefetch_b8)
                __builtin_prefetch(Ag + (size_t)a_row * lda + k0 + 2 * BK + a_col, 0, 0);
                __builtin_prefetch(Bg + (size_t)(k0 + 2 * BK + 2 * b_kp) * ldb + b_col, 0, 0);
            }
        }

        #pragma unroll
        for (int kk = 0; kk < BK; kk += 4) {
            // A fragment (16x4 f32): lanes 0-15 -> K{kk,kk+1}, lanes 16-31 -> K{kk+2,kk+3}, row = lr
            v2f a[4];
            #pragma unroll
            for (int i = 0; i < 4; ++i) {
                const float* p = As + (wm * 64 + i * 16 + lr) * LDA_S + kk + 2 * lh;
                a[i].x = p[0];
                a[i].y = p[1];
            }
            // B fragment (4x16 f32): one b64 load of the interleaved pair {B[k][n], B[k+1][n]}
            v2f b[2];
            #pragma unroll
            for (int j = 0; j < 2; ++j) {
                const float* p = Bs + ((kk >> 1) + lh) * LDB_P + 2 * (wn * 32 + j * 16 + lr);
                b[j].x = p[0];
                b[j].y = p[1];
            }
            #pragma unroll
            for (int i = 0; i < 4; ++i)
                #pragma unroll
                for (int j = 0; j < 2; ++j)
                    acc[i][j] = __builtin_amdgcn_wmma_f32_16x16x4_f32(
                        false, a[i], false, b[j], (short)0, acc[i][j], false, false);
        }
        __syncthreads();
        if (has_next) {
            store_tile();
            __syncthreads();
        }
    }
}

// Kernel 1: S[8192,1024] = Yflat[8192,1024] @ W[1024,1024]  (batch folded into M, W shared)
__global__ void __launch_bounds__(256) gcn_support_kernel(
    const float* __restrict__ Y, const float* __restrict__ W,
    float* __restrict__ S, int Mtot, int K, int N)
{
    __shared__ float As[BM * LDA_S];
    __shared__ float Bs[(BK / 2) * LDB_P];

    v8f acc[4][2];
    #pragma unroll
    for (int i = 0; i < 4; ++i)
        #pragma unroll
        for (int j = 0; j < 2; ++j)
            #pragma unroll
            for (int r = 0; r < 8; ++r)
                acc[i][j][r] = 0.0f;

    const int tileN = blockIdx.x * BN;
    const int tileM = blockIdx.y * BM;

    gemm_mainloop(Y, W, K, N, K, tileM, tileN, As, Bs, acc);

    const int lane = threadIdx.x & 31;
    const int wave = threadIdx.x >> 5;
    const int wm = wave >> 2, wn = wave & 3;
    const int lr = lane & 15, lh = lane >> 4;

    #pragma unroll
    for (int i = 0; i < 4; ++i)
        #pragma unroll
        for (int j = 0; j < 2; ++j) {
            const int col = tileN + wn * 32 + j * 16 + lr;
            #pragma unroll
            for (int r = 0; r < 8; ++r) {
                const int row = tileM + wm * 64 + i * 16 + r + 8 * lh;
                S[(size_t)row * N + col] = acc[i][j][r];
            }
        }
}

// Kernel 2 (per batch z): Out[b] = relu( (Adj[b] @ S[b]) / sumrow[b] + bias + X[b] )
__global__ void __launch_bounds__(256) gcn_aggregate_kernel(
    const float* __restrict__ Adj, const float* __restrict__ S,
    const float* __restrict__ sumrow, const float* __restrict__ bias,
    const float* __restrict__ X, float* __restrict__ Out,
    int Nn, int Dd)
{
    __shared__ float As[BM * LDA_S];
    __shared__ float Bs[(BK / 2) * LDB_P];

    const int bz = blockIdx.z;
    const float* A  = Adj + (size_t)bz * Nn * Nn;
    const float* Bm = S   + (size_t)bz * Nn * Dd;
    const float* Xb = X   + (size_t)bz * Nn * Dd;
    float*       Ob = Out + (size_t)bz * Nn * Dd;
    const float* sr = sumrow + (size_t)bz * Nn;

    v8f acc[4][2];
    #pragma unroll
    for (int i = 0; i < 4; ++i)
        #pragma unroll
        for (int j = 0; j < 2; ++j)
            #pragma unroll
            for (int r = 0; r < 8; ++r)
                acc[i][j][r] = 0.0f;

    const int tileN = blockIdx.x * BN;
    const int tileM = blockIdx.y * BM;

    gemm_mainloop(A, Bm, Nn, Dd, Nn, tileM, tileN, As, Bs, acc);

    const int lane = threadIdx.x & 31;
    const int wave = threadIdx.x >> 5;
    const int wm = wave >> 2, wn = wave & 3;
    const int lr = lane & 15, lh = lane >> 4;

    #pragma unroll
    for (int i = 0; i < 4; ++i) {
        const int rowbase = tileM + wm * 64 + i * 16 + 8 * lh;
        float inv[8];
        #pragma unroll
        for (int r = 0; r < 8; ++r)
            inv[r] = 1.0f / sr[rowbase + r];
        #pragma unroll
        for (int j = 0; j < 2; ++j) {
            const int col = tileN + wn * 32 + j * 16 + lr;
            const float bb = bias[col];
            #pragma unroll
            for (int r = 0; r < 8; ++r) {
                const int row = rowbase + r;
                float v = acc[i][j][r] * inv[r] + bb + Xb[(size_t)row * Dd + col];
                Ob[(size_t)row * Dd + col] = fmaxf(v, 0.0f);
            }
        }
    }
}

extern "C" void kernel_launch(void* const* d_in, const int* in_sizes, int n_in,
                              void* d_out, int out_size, void* d_ws, size_t ws_size,
                              hipStream_t stream) {
    const int Bb = 8, Nn = 1024, Dd = 1024;

    const float* x    = (const float*)d_in[0];
    const float* y    = (const float*)d_in[1];
    const float* adj  = (const float*)d_in[2];
    const float* srow = (const float*)d_in[3];
    const float* W    = (const float*)d_in[4];
    const float* bias = (const float*)d_in[5];
    float* out = (float*)d_out;
    float* S   = (float*)d_ws;   // needs B*N*D*4 = 32 MiB scratch

    dim3 blk(256);
    // support = y @ W : fold batch into M (W shared across batch)
    dim3 g1(Dd / BN, (Bb * Nn) / BM);
    gcn_support_kernel<<<g1, blk, 0, stream>>>(y, W, S, Bb * Nn, Dd, Dd);
    // out = relu(adj @ support / sumrow + bias + x), batched over z
    dim3 g2(Dd / BN, Nn / BM, Bb);
    gcn_aggregate_kernel<<<g2, blk, 0, stream>>>(adj, S, srow, bias, x, out, Nn, Dd);
}